// SpectralGraphEncoder_87093346828563
// MI455X (gfx1250) — compile-verified
//
#include <hip/hip_runtime.h>
#include <math.h>

// ---------------------------------------------------------------------------
// Spectral graph encoder for MI455X (gfx1250, wave32).
// eigh(L) restructured as WMMA-driven subspace iteration on
// M = D^-1/2 A D^-1/2  (top-17 subspace == bottom-17 of L).
// Hot loop: Z = A * P, P = D^-1/2 Q staged in LDS, via V_WMMA_F32_16X16X4_F32.
// Inner loop is load/load/wmma only: all diagonal scaling hoisted out.
// ---------------------------------------------------------------------------

typedef __attribute__((ext_vector_type(2))) float v2f;
typedef __attribute__((ext_vector_type(8))) float v8f;

#define NN      256   // nodes per graph
#define SDIM    17    // subspace size (need eigvecs 1..16 -> top 17 of M)
#define QSTR    20    // Qs LDS row stride (17 live cols + pad)
#define PSTR    32    // Ps LDS row stride (2 x 16-wide WMMA col tiles)
#define NITER   30    // orthogonal-iteration count
#define JSWEEPS 12    // Jacobi sweeps on the 17x17 Ritz matrix

// Block-wide sum over 256 threads via LDS tree. Leading barrier protects the
// scratch buffer from the previous use.
__device__ __forceinline__ float block_reduce_sum(float v, float* red, int tid) {
  __syncthreads();
  red[tid] = v;
  __syncthreads();
#pragma unroll
  for (int s = 128; s > 0; s >>= 1) {
    if (tid < s) red[tid] += red[tid + s];
    __syncthreads();
  }
  return red[0];
}

// One application of M to the basis:
//   stage  P = D^-1/2 Q                       (LDS, one mul/element)
//   GEMM   ACC = A * P                        (pure load+wmma inner loop)
//   write  Z = D^-1/2 ACC [+ Q]  -> Qs (shift) or Ps (Rayleigh-Ritz Y)
// 8 waves x 4 tiles each = 32 16x16 output tiles, K=256 in steps of 4.
__device__ __forceinline__ void spectral_gemm(const float* __restrict__ Ab,
                                              const float* __restrict__ dinv,
                                              float* __restrict__ Qs,
                                              float* __restrict__ Ps,
                                              bool shift_inplace, int tid) {
  // ---- stage P = D^-1/2 Q (pad cols of Ps stay zero) -----------------------
  {
    const float dr = dinv[tid];
#pragma unroll
    for (int s = 0; s < SDIM; ++s)
      Ps[tid * PSTR + s] = dr * Qs[tid * QSTR + s];
  }
  __syncthreads();

  const int wave = tid >> 5;
  const int lane = tid & 31;
  const int half = lane >> 4;   // A/B fragments: lanes 16-31 hold K+2,K+3
  const int lr   = lane & 15;

  v8f accs[4];
#pragma unroll
  for (int t = 0; t < 4; ++t) {
    const int tile = wave * 4 + t;      // 0..31
    const int rt   = tile & 15;         // row tile
    const int ct   = tile >> 4;         // col tile
    const int arow = rt * 16 + lr;      // A-fragment row for this lane
    const int bcol = ct * 16 + lr;      // B-fragment column for this lane
    const float* Arow = Ab + arow * NN;
    __builtin_prefetch(Arow + 128, 0, 0);   // global_prefetch_b8: 2nd half of row

    v8f acc = {0.f, 0.f, 0.f, 0.f, 0.f, 0.f, 0.f, 0.f};
#pragma unroll 8
    for (int kb = 0; kb < NN; kb += 4) {
      const int ka = kb + 2 * half;
      const float2 av = *reinterpret_cast<const float2*>(Arow + ka);
      v2f a;
      a.x = av.x;
      a.y = av.y;
      v2f b;
      b.x = Ps[ka * PSTR + bcol];
      b.y = Ps[(ka + 1) * PSTR + bcol];
      acc = __builtin_amdgcn_wmma_f32_16x16x4_f32(false, a, false, b,
                                                  (short)0, acc, false, false);
    }
    accs[t] = acc;
  }

  __syncthreads();  // all waves done reading Ps before Ps/Qs writeback

#pragma unroll
  for (int t = 0; t < 4; ++t) {
    const int tile = wave * 4 + t;
    const int rt   = tile & 15;
    const int ct   = tile >> 4;
    const int gcol = ct * 16 + lr;
    if (gcol < SDIM) {
#pragma unroll
      for (int v = 0; v < 8; ++v) {
        const int grow = rt * 16 + v + 8 * half;  // C layout: lanes>=16 hold M+8
        const float z = dinv[grow] * accs[t][v];
        if (shift_inplace) {
          // Z = M*Q + Q (shift to [0,2] keeps top-eigen ordering)
          Qs[grow * QSTR + gcol] += z;
        } else {
          Ps[grow * PSTR + gcol] = z;   // Rayleigh-Ritz Y, in place
        }
      }
    }
  }
  __syncthreads();
}

// Modified Gram-Schmidt over the SDIM live columns (thread t owns node t).
__device__ __forceinline__ void mgs(float* __restrict__ Qs, float* __restrict__ red,
                                    int tid) {
  for (int j = 0; j < SDIM; ++j) {
    float qj  = Qs[tid * QSTR + j];
    float nrm = block_reduce_sum(qj * qj, red, tid);
    qj *= rsqrtf(nrm + 1e-24f);
    Qs[tid * QSTR + j] = qj;
    __syncthreads();
    for (int i = j + 1; i < SDIM; ++i) {
      float qi = Qs[tid * QSTR + i];
      float d  = block_reduce_sum(qj * qi, red, tid);
      Qs[tid * QSTR + i] = qi - d * qj;
      __syncthreads();
    }
  }
}

__global__ __launch_bounds__(256, 1)
void spectral_encoder_kernel(const float* __restrict__ adj,
                             const float* __restrict__ W,
                             const float* __restrict__ bvec,
                             float* __restrict__ out) {
  __shared__ float Qs[NN * QSTR];      // 20 KB  orthonormal basis
  __shared__ float Ps[NN * PSTR];      // 32 KB  D^-1/2 Q (GEMM operand) / RR Y
  __shared__ float red[NN];            //  1 KB  reduction scratch
  __shared__ float dinv[NN];           //  1 KB  d^-1/2
  __shared__ float Tm[SDIM * SDIM];    //  Ritz matrix
  __shared__ float Um[SDIM * SDIM];    //  Ritz rotations
  __shared__ float evs[SDIM];
  __shared__ int   idxs[SDIM];
  __shared__ float qm[SDIM];           //  column means of Q
  __shared__ float emb[16];
  __shared__ float proj[16];
  __shared__ float snorm;

  const int tid = threadIdx.x;
  const int g   = blockIdx.x;
  const float* __restrict__ Ab = adj + (size_t)g * NN * NN;

  // ---- degree and d^-1/2 (wave-parallel row sums, coalesced across lanes) --
  {
    const int wave = tid >> 5;
    const int lane = tid & 31;
    for (int r = wave * 32; r < wave * 32 + 32; ++r) {
      float s = 0.f;
#pragma unroll
      for (int m = 0; m < 8; ++m) s += Ab[r * NN + lane + 32 * m];
#pragma unroll
      for (int off = 16; off > 0; off >>= 1) s += __shfl_xor(s, off);
      if (lane == 0) dinv[r] = rsqrtf(s + 1e-8f);
    }
  }

  // ---- deterministic pseudo-random init of Q; zero Ps pad columns ----------
#pragma unroll
  for (int s = 0; s < QSTR; ++s) {
    float v = 0.f;
    if (s < SDIM) {
      unsigned h = (unsigned)tid * 2654435761u + (unsigned)s * 40503u + 0x9e3779b9u;
      h ^= h >> 16; h *= 0x85ebca6bu; h ^= h >> 13; h *= 0xc2b2ae35u; h ^= h >> 16;
      v = ((h & 0xFFFFFFu) * (1.0f / 8388608.0f)) - 1.0f;
    }
    Qs[tid * QSTR + s] = v;
  }
#pragma unroll
  for (int s = SDIM; s < PSTR; ++s) Ps[tid * PSTR + s] = 0.f;
  __syncthreads();
  mgs(Qs, red, tid);

  // ---- orthogonal iteration: Q <- orth((M + I) Q) --------------------------
  for (int it = 0; it < NITER; ++it) {
    spectral_gemm(Ab, dinv, Qs, Ps, /*shift_inplace=*/true, tid);
    mgs(Qs, red, tid);
  }

  // ---- Rayleigh-Ritz: Y = M*Q (into Ps), T = Q^T Y -------------------------
  spectral_gemm(Ab, dinv, Qs, Ps, /*shift_inplace=*/false, tid);
  for (int p = 0; p < SDIM * SDIM; ++p) {
    const int i = p / SDIM, j = p % SDIM;
    float d = block_reduce_sum(Qs[tid * QSTR + i] * Ps[tid * PSTR + j], red, tid);
    if (tid == 0) Tm[p] = d;
  }
  __syncthreads();

  // ---- serial cyclic Jacobi on the 17x17 Ritz matrix (thread 0) ------------
  if (tid == 0) {
    for (int i = 0; i < SDIM; ++i)
      for (int j = i + 1; j < SDIM; ++j) {
        float v = 0.5f * (Tm[i * SDIM + j] + Tm[j * SDIM + i]);
        Tm[i * SDIM + j] = v; Tm[j * SDIM + i] = v;
      }
    for (int i = 0; i < SDIM; ++i)
      for (int j = 0; j < SDIM; ++j)
        Um[i * SDIM + j] = (i == j) ? 1.f : 0.f;

    for (int sweep = 0; sweep < JSWEEPS; ++sweep) {
      for (int p = 0; p < SDIM - 1; ++p) {
        for (int q = p + 1; q < SDIM; ++q) {
          float apq = Tm[p * SDIM + q];
          if (fabsf(apq) < 1e-12f) continue;
          float app = Tm[p * SDIM + p];
          float aqq = Tm[q * SDIM + q];
          float theta = 0.5f * (aqq - app) / apq;
          float tsgn  = (theta >= 0.f) ? 1.f : -1.f;
          float tt    = tsgn / (fabsf(theta) + sqrtf(theta * theta + 1.f));
          float c     = rsqrtf(tt * tt + 1.f);
          float s     = tt * c;
          for (int k = 0; k < SDIM; ++k) {
            float akp = Tm[k * SDIM + p], akq = Tm[k * SDIM + q];
            Tm[k * SDIM + p] = c * akp - s * akq;
            Tm[k * SDIM + q] = s * akp + c * akq;
          }
          for (int k = 0; k < SDIM; ++k) {
            float apk = Tm[p * SDIM + k], aqk = Tm[q * SDIM + k];
            Tm[p * SDIM + k] = c * apk - s * aqk;
            Tm[q * SDIM + k] = s * apk + c * aqk;
          }
          for (int k = 0; k < SDIM; ++k) {
            float up = Um[k * SDIM + p], uq = Um[k * SDIM + q];
            Um[k * SDIM + p] = c * up - s * uq;
            Um[k * SDIM + q] = s * up + c * uq;
          }
        }
      }
    }
    // sort eigenvalues of M descending == eigenvalues of L ascending
    for (int i = 0; i < SDIM; ++i) { evs[i] = Tm[i * SDIM + i]; idxs[i] = i; }
    for (int i = 0; i < SDIM - 1; ++i) {
      int best = i;
      for (int j = i + 1; j < SDIM; ++j)
        if (evs[idxs[j]] > evs[idxs[best]]) best = j;
      int tmp = idxs[i]; idxs[i] = idxs[best]; idxs[best] = tmp;
    }
  }
  __syncthreads();

  // ---- column means of Q (node-mean commutes with V = Q*U) -----------------
  for (int s = 0; s < SDIM; ++s) {
    float d = block_reduce_sum(Qs[tid * QSTR + s], red, tid);
    if (tid == 0) qm[s] = d * (1.0f / NN);
  }
  __syncthreads();

  // ---- embedding (skip Ritz pair 0), linear layer, Poincare projection -----
  if (tid < 16) {
    const int col = idxs[tid + 1];   // L-eigenvector indices 1..16
    float e = 0.f;
    for (int s = 0; s < SDIM; ++s) e += qm[s] * Um[s * SDIM + col];
    emb[tid] = e;
  }
  __syncthreads();
  if (tid < 16) {
    float p = bvec[tid];
    for (int s = 0; s < 16; ++s) p += W[tid * 16 + s] * emb[s];
    proj[tid] = p;
  }
  __syncthreads();
  if (tid == 0) {
    float n2 = 0.f;
    for (int s = 0; s < 16; ++s) n2 += proj[s] * proj[s];
    const float nrm   = sqrtf(n2);
    const float max_r = 0.95f;  // MAX_NORM / sqrt(CURVATURE)
    snorm = fminf(1.0f, max_r / (nrm + 1e-8f));
  }
  __syncthreads();
  if (tid < 16) out[g * 16 + tid] = proj[tid] * snorm;
}

extern "C" void kernel_launch(void* const* d_in, const int* in_sizes, int n_in,
                              void* d_out, int out_size, void* d_ws, size_t ws_size,
                              hipStream_t stream) {
  (void)in_sizes; (void)n_in; (void)d_ws; (void)ws_size; (void)out_size;
  const float* adj = (const float*)d_in[0];   // (128,256,256) fp32
  const float* W   = (const float*)d_in[1];   // (16,16) fp32
  const float* b   = (const float*)d_in[2];   // (16,)  fp32
  float* out = (float*)d_out;                 // (128,16) fp32
  hipLaunchKernelGGL(spectral_encoder_kernel, dim3(128), dim3(256), 0, stream,
                     adj, W, b, out);
}